// GatedGCNGraphGymLayer_54503134986741
// MI455X (gfx1250) — compile-verified
//
#include <hip/hip_runtime.h>
#include <math.h>

#define DIM 128

typedef __attribute__((ext_vector_type(2))) float v2f;
typedef __attribute__((ext_vector_type(8))) float v8f;

// ---------------------------------------------------------------------------
// Kernel 0: zero the accumulation regions (num, den, BN stats)
// ---------------------------------------------------------------------------
__global__ __launch_bounds__(256) void zero_kernel(float* __restrict__ p, long n) {
    long i = (long)blockIdx.x * blockDim.x + threadIdx.x;
    if (i < n) p[i] = 0.0f;
}

// ---------------------------------------------------------------------------
// Kernel 1: fused 4-way GEMM  Ax/Bx/Dx/Ex = x @ {Wa,Wb,Wd,We} + bias
// One block = 16-node row strip; 8 waves, wave w owns columns [16w,16w+16).
// Each wave keeps four v8f accumulators and reuses the A fragment of x
// across the four weight matrices.  V_WMMA_F32_16X16X4_F32, 32 K-steps.
// Epilogue: single uniform full-tile check -> unguarded stores on the
// common path (no per-element exec-mask juggling).
// ---------------------------------------------------------------------------
__global__ __launch_bounds__(256) void gemm4_wmma_kernel(
    const float* __restrict__ x,
    const float* __restrict__ Wa, const float* __restrict__ ba,
    const float* __restrict__ Wb, const float* __restrict__ bb,
    const float* __restrict__ Wd, const float* __restrict__ bd,
    const float* __restrict__ We, const float* __restrict__ be,
    float* __restrict__ Ax, float* __restrict__ Bx,
    float* __restrict__ Dx, float* __restrict__ Ex,
    int n_nodes)
{
    const int lane = threadIdx.x & 31;
    const int wave = threadIdx.x >> 5;
    const int l16  = lane & 15;
    const int half = lane >> 4;          // 0: lanes 0-15, 1: lanes 16-31
    const int row0 = blockIdx.x * 16;
    const int col0 = wave * 16;
    const int col  = col0 + l16;

    // A-fragment source row (clamped so loads stay in range; no EXEC divergence)
    int arow = row0 + l16;
    if (arow >= n_nodes) arow = n_nodes - 1;
    const float* __restrict__ xrow = x + (size_t)arow * DIM;

    const float* __restrict__ Ws[4] = { Wa, Wb, Wd, We };
    const float* __restrict__ bs[4] = { ba, bb, bd, be };
    float* __restrict__ outs[4]     = { Ax, Bx, Dx, Ex };

    v8f acc[4] = {};

    #pragma unroll 4
    for (int k = 0; k < DIM / 4; ++k) {
        const int kk = 4 * k + 2 * half;     // per-lane K origin for this step
        v2f a;
        a.x = xrow[kk];
        a.y = xrow[kk + 1];
        #pragma unroll
        for (int m = 0; m < 4; ++m) {
            v2f b;
            b.x = Ws[m][(size_t)kk * DIM + col];
            b.y = Ws[m][(size_t)(kk + 1) * DIM + col];
            acc[m] = __builtin_amdgcn_wmma_f32_16x16x4_f32(
                /*neg_a=*/false, a, /*neg_b=*/false, b,
                /*c_mod=*/(short)0, acc[m],
                /*reuse_a=*/false, /*reuse_b=*/false);
        }
    }

    // C layout: VGPR j -> row (row0 + j + 8*half), col (col0 + l16)
    if (row0 + 16 <= n_nodes) {
        // Fast path: full tile, straight-line stores (clause-friendly).
        #pragma unroll
        for (int m = 0; m < 4; ++m) {
            const float bias = bs[m][col];
            float* __restrict__ op = outs[m] + (size_t)(row0 + 8 * half) * DIM + col;
            #pragma unroll
            for (int j = 0; j < 8; ++j) {
                op[(size_t)j * DIM] = acc[m][j] + bias;
            }
        }
    } else {
        // Tail tile only.
        #pragma unroll
        for (int m = 0; m < 4; ++m) {
            const float bias = bs[m][col];
            #pragma unroll
            for (int j = 0; j < 8; ++j) {
                const int row = row0 + j + 8 * half;
                if (row < n_nodes)
                    outs[m][(size_t)row * DIM + col] = acc[m][j] + bias;
            }
        }
    }
}

// ---------------------------------------------------------------------------
// Kernel 2: edge phase.  One wave32 per edge, float4 per lane (128 dims).
//   e_ij = Dx[dst] + Ex[src]           -> written to output region
//   sigma = sigmoid(e_ij)
//   num[dst] += sigma * Bx[src]        (global fp32 atomics)
//   den[dst] += sigma
// ---------------------------------------------------------------------------
__global__ __launch_bounds__(256) void edge_kernel(
    const int* __restrict__ src, const int* __restrict__ dst,
    const float* __restrict__ Bx, const float* __restrict__ Dx,
    const float* __restrict__ Ex,
    float* __restrict__ e_out, float* __restrict__ num,
    float* __restrict__ den, int n_edges)
{
    const int edge = (int)((blockIdx.x * (unsigned)blockDim.x + threadIdx.x) >> 5);
    const int lane = threadIdx.x & 31;
    if (edge >= n_edges) return;

    const int s = src[edge];
    const int d = dst[edge];

    const float4 dv = ((const float4*)(Dx + (size_t)d * DIM))[lane];
    const float4 ev = ((const float4*)(Ex + (size_t)s * DIM))[lane];
    const float4 bv = ((const float4*)(Bx + (size_t)s * DIM))[lane];

    float4 eij;
    eij.x = dv.x + ev.x; eij.y = dv.y + ev.y;
    eij.z = dv.z + ev.z; eij.w = dv.w + ev.w;
    ((float4*)(e_out + (size_t)edge * DIM))[lane] = eij;

    const float sx = 1.0f / (1.0f + __expf(-eij.x));
    const float sy = 1.0f / (1.0f + __expf(-eij.y));
    const float sz = 1.0f / (1.0f + __expf(-eij.z));
    const float sw = 1.0f / (1.0f + __expf(-eij.w));

    float* np = num + (size_t)d * DIM + 4 * lane;
    float* dp = den + (size_t)d * DIM + 4 * lane;
    atomicAdd(np + 0, sx * bv.x);
    atomicAdd(np + 1, sy * bv.y);
    atomicAdd(np + 2, sz * bv.z);
    atomicAdd(np + 3, sw * bv.w);
    atomicAdd(dp + 0, sx);
    atomicAdd(dp + 1, sy);
    atomicAdd(dp + 2, sz);
    atomicAdd(dp + 3, sw);
}

// ---------------------------------------------------------------------------
// Kernel 3: h = Ax + num/(den+eps)  (in-place over Ax) + per-channel BN
// partial sums.  128 threads/block, thread t owns channel t (coalesced),
// grid-strided over nodes, two atomics per thread at the end.
// ---------------------------------------------------------------------------
__global__ __launch_bounds__(128) void h_stats_kernel(
    float* __restrict__ h /* == Ax region */,
    const float* __restrict__ num, const float* __restrict__ den,
    float* __restrict__ ssum, float* __restrict__ ssq, int n_nodes)
{
    const int c = threadIdx.x;   // channel 0..127
    float s = 0.0f, q = 0.0f;
    for (int n = blockIdx.x; n < n_nodes; n += gridDim.x) {
        const size_t idx = (size_t)n * DIM + c;
        const float hv = h[idx] + num[idx] / (den[idx] + 1e-6f);
        h[idx] = hv;
        s += hv;
        q += hv * hv;
    }
    atomicAdd(&ssum[c], s);
    atomicAdd(&ssq[c], q);
}

// ---------------------------------------------------------------------------
// Kernel 4: BatchNorm (training stats, biased var) + affine + ReLU
// ---------------------------------------------------------------------------
__global__ __launch_bounds__(256) void norm_kernel(
    const float* __restrict__ h,
    const float* __restrict__ ssum, const float* __restrict__ ssq,
    const float* __restrict__ gamma, const float* __restrict__ beta,
    float* __restrict__ x_out, int n_nodes)
{
    const long i = (long)blockIdx.x * blockDim.x + threadIdx.x;
    const long total = (long)n_nodes * DIM;
    if (i >= total) return;
    const int c = (int)(i & (DIM - 1));
    const float inv_n = 1.0f / (float)n_nodes;
    const float mean  = ssum[c] * inv_n;
    const float var   = ssq[c] * inv_n - mean * mean;
    const float rstd  = rsqrtf(var + 1e-5f);
    const float v = gamma[c] * ((h[i] - mean) * rstd) + beta[c];
    x_out[i] = fmaxf(v, 0.0f);
}

// ---------------------------------------------------------------------------
// Launcher
// ---------------------------------------------------------------------------
extern "C" void kernel_launch(void* const* d_in, const int* in_sizes, int n_in,
                              void* d_out, int out_size, void* d_ws, size_t ws_size,
                              hipStream_t stream) {
    const float* x     = (const float*)d_in[0];
    // d_in[1] (e) is unused by the reference forward
    const int*   eidx  = (const int*)d_in[2];
    const float* Wa    = (const float*)d_in[3];
    const float* ba    = (const float*)d_in[4];
    const float* Wb    = (const float*)d_in[5];
    const float* bb    = (const float*)d_in[6];
    const float* Wd    = (const float*)d_in[7];
    const float* bd    = (const float*)d_in[8];
    const float* We    = (const float*)d_in[9];
    const float* be    = (const float*)d_in[10];
    const float* gamma = (const float*)d_in[11];
    const float* beta  = (const float*)d_in[12];

    const int n_nodes = in_sizes[0] / DIM;
    const int n_edges = in_sizes[2] / 2;
    const int* src = eidx;             // edge_index[0]
    const int* dst = eidx + n_edges;   // edge_index[1]

    const size_t NN = (size_t)n_nodes * DIM;   // 12.8M floats

    float* ws   = (float*)d_ws;
    float* Ax   = ws;            // also becomes h in-place
    float* Bx   = ws + 1 * NN;
    float* Dx   = ws + 2 * NN;
    float* Ex   = ws + 3 * NN;
    float* numb = ws + 4 * NN;
    float* denb = ws + 5 * NN;
    float* ssum = ws + 6 * NN;
    float* ssq  = ssum + DIM;

    float* x_out = (float*)d_out;
    float* e_out = (float*)d_out + NN;

    // 0) zero num, den, stats
    {
        const long nz = (long)(2 * NN) + 2 * DIM;
        const int blocks = (int)((nz + 255) / 256);
        zero_kernel<<<blocks, 256, 0, stream>>>(numb, nz);
    }

    // 1) fused 4-way WMMA GEMM
    {
        const int row_tiles = (n_nodes + 15) / 16;   // 6250
        gemm4_wmma_kernel<<<row_tiles, 256, 0, stream>>>(
            x, Wa, ba, Wb, bb, Wd, bd, We, be, Ax, Bx, Dx, Ex, n_nodes);
    }

    // 2) edge gather / gate / scatter (one wave per edge)
    {
        const int waves_per_block = 256 / 32;
        const int blocks = (n_edges + waves_per_block - 1) / waves_per_block;
        edge_kernel<<<blocks, 256, 0, stream>>>(
            src, dst, Bx, Dx, Ex, e_out, numb, denb, n_edges);
    }

    // 3) h = Ax + aggr (in place) + BN partial sums
    h_stats_kernel<<<1024, 128, 0, stream>>>(Ax, numb, denb, ssum, ssq, n_nodes);

    // 4) normalize + affine + ReLU
    {
        const long total = (long)n_nodes * DIM;
        const int blocks = (int)((total + 255) / 256);
        norm_kernel<<<blocks, 256, 0, stream>>>(Ax, ssum, ssq, gamma, beta, x_out, n_nodes);
    }
}